// KNNMemoryAttention_80401787781788
// MI455X (gfx1250) — compile-verified
//
#include <hip/hip_runtime.h>
#include <stdint.h>

// KNN-memory attention for MI455X (gfx1250, wave32).
// Memory-bound: 512MB of mem_k/mem_v streamed once -> ~23us floor @ 23.3TB/s.
// fp32 WMMA (V_WMMA_F32_16X16X4_F32) does the dot products at reference precision.

typedef __attribute__((ext_vector_type(2))) float v2f;
typedef __attribute__((ext_vector_type(8))) float v8f;

namespace {
constexpr int Bb = 2, Nn = 2048, Hh = 8, Dd = 64, Kk = 32;
constexpr int WAVES = 8;                  // waves per block
constexpr int NQ = Bb * Hh * Nn;          // 32768 queries
constexpr float NEG_MAX = -3.402823466e+38f;
}

__global__ __launch_bounds__(WAVES * 32)
void knn_mem_attn_kernel(const float* __restrict__ q,
                         const float* __restrict__ mem_k,
                         const float* __restrict__ mem_v,
                         const uint8_t* __restrict__ mem_mask,   // jnp.bool_: 1 byte/elem
                         const float* __restrict__ local_out,
                         const float* __restrict__ scale,
                         float* __restrict__ out)
{
    __shared__ float lds_attn[WAVES][Kk];

    const int lane = threadIdx.x & 31;
    const int wave = threadIdx.x >> 5;
    const int half = lane >> 4;          // 0: lanes 0-15, 1: lanes 16-31
    const int lrow = lane & 15;          // M (A) / N (B,C) index within a tile
    const int kin  = half << 1;          // this lane's K sub-pair base (0 or 2)
    const int base = half << 3;          // first C-row (M) this lane holds

    // qid enumerates (b,h,n) in mem_k's layout order: ((b*H + h)*N + n)
    const int qid = blockIdx.x * WAVES + wave;
    const int b = qid >> 14;             // / (H*N)
    const int h = (qid >> 11) & (Hh - 1);
    const int n = qid & (Nn - 1);

    const float*   kp = mem_k + (size_t)qid * (Kk * Dd);
    const float*   vp = mem_v + (size_t)qid * (Kk * Dd);
    const uint8_t* mp = mem_mask + (size_t)qid * Kk;
    const size_t qoff = ((size_t)(b * Nn + n) * Hh + h) * Dd;  // q/local_out/out offset
    const float*   qp = q + qoff;

    // ---------------- sim = mem_k (32x64) . q (64) ----------------
    // Two 16x16 WMMA tiles (j=0..15 and j=16..31), K chained over D=64 in 16 steps.
    // A[m][k] = mem_k[tile*16+m][4c+k]; B[k][n] = q[4c+k] (replicated over N).
    v8f c0 = {0.f,0.f,0.f,0.f,0.f,0.f,0.f,0.f};
    v8f c1 = c0;
#pragma unroll
    for (int c = 0; c < 16; ++c) {
        const int d0 = 4 * c + kin;
        v2f bq = *(const v2f*)(qp + d0);  // wave-uniform per half; L1 hit
        v2f a0 = __builtin_nontemporal_load((const v2f*)(kp + (size_t)lrow * Dd + d0));
        v2f a1 = __builtin_nontemporal_load((const v2f*)(kp + (size_t)(16 + lrow) * Dd + d0));
        c0 = __builtin_amdgcn_wmma_f32_16x16x4_f32(false, a0, false, bq, (short)0, c0, false, false);
        c1 = __builtin_amdgcn_wmma_f32_16x16x4_f32(false, a1, false, bq, (short)0, c1, false, false);
    }

    // ---------------- scale, mask, softmax over K=32 ----------------
    // Lane holds sim[j] for j = base..base+7 (c0) and 16+base..16+base+7 (c1).
    const float s = expf(scale[h]);
    const uint64_t mb0 = *(const uint64_t*)(mp + base);
    const uint64_t mb1 = *(const uint64_t*)(mp + 16 + base);

    float x0[8], x1[8];
    float mx = NEG_MAX;
#pragma unroll
    for (int r = 0; r < 8; ++r) {
        x0[r] = ((mb0 >> (8 * r)) & 0xffu) ? c0[r] * s : NEG_MAX;
        x1[r] = ((mb1 >> (8 * r)) & 0xffu) ? c1[r] * s : NEG_MAX;
        mx = fmaxf(mx, fmaxf(x0[r], x1[r]));
    }
    mx = fmaxf(mx, __shfl_xor(mx, 16, 32));   // lane-pair (l, l^16) covers all 32 j

    float p0[8], p1[8];
    float sum = 0.f;
#pragma unroll
    for (int r = 0; r < 8; ++r) {
        p0[r] = expf(x0[r] - mx);
        p1[r] = expf(x1[r] - mx);
        sum += p0[r] + p1[r];
    }
    sum += __shfl_xor(sum, 16, 32);
    const float inv = 1.f / sum;

    // Redistribute attn through LDS (lanes 0 and 16 jointly hold all 32 values).
    if (lrow == 0) {
#pragma unroll
        for (int r = 0; r < 8; ++r) {
            lds_attn[wave][base + r]      = p0[r] * inv;
            lds_attn[wave][16 + base + r] = p1[r] * inv;
        }
    }
    __syncthreads();   // all 8 waves execute identical control flow

    // ---------------- mem_out = attn (32) . mem_v (32x64) ----------------
    // 4 d-tiles of 16; A[m][k] = attn[4c2+k] (replicated over M);
    // B[k][n] = mem_v[4c2+k][tile*16+n]. K chained over 32 in 8 steps.
    v8f o0 = {0.f,0.f,0.f,0.f,0.f,0.f,0.f,0.f};
    v8f o1 = o0, o2 = o0, o3 = o0;
    const float* la = lds_attn[wave];
#pragma unroll
    for (int c2 = 0; c2 < 8; ++c2) {
        const int j0 = 4 * c2 + kin;
        v2f aa = *(const v2f*)(la + j0);
        const float* v0r = vp + (size_t)j0 * Dd + lrow;
        const float* v1r = vp + (size_t)(j0 + 1) * Dd + lrow;
        v2f b0; b0.x = __builtin_nontemporal_load(v0r + 0);  b0.y = __builtin_nontemporal_load(v1r + 0);
        v2f b1; b1.x = __builtin_nontemporal_load(v0r + 16); b1.y = __builtin_nontemporal_load(v1r + 16);
        v2f b2; b2.x = __builtin_nontemporal_load(v0r + 32); b2.y = __builtin_nontemporal_load(v1r + 32);
        v2f b3; b3.x = __builtin_nontemporal_load(v0r + 48); b3.y = __builtin_nontemporal_load(v1r + 48);
        o0 = __builtin_amdgcn_wmma_f32_16x16x4_f32(false, aa, false, b0, (short)0, o0, false, false);
        o1 = __builtin_amdgcn_wmma_f32_16x16x4_f32(false, aa, false, b1, (short)0, o1, false, false);
        o2 = __builtin_amdgcn_wmma_f32_16x16x4_f32(false, aa, false, b2, (short)0, o2, false, false);
        o3 = __builtin_amdgcn_wmma_f32_16x16x4_f32(false, aa, false, b3, (short)0, o3, false, false);
    }

    // C2[m][n] = out[tile*16+n] for every m; columns duplicate across halves,
    // so only lanes 0-15 store. Fuse the local_out residual add.
    if (half == 0) {
        const float* lp = local_out + qoff;
        float*       op = out + qoff;
        op[lrow]      = lp[lrow]      + o0[0];
        op[16 + lrow] = lp[16 + lrow] + o1[0];
        op[32 + lrow] = lp[32 + lrow] + o2[0];
        op[48 + lrow] = lp[48 + lrow] + o3[0];
    }
}

extern "C" void kernel_launch(void* const* d_in, const int* in_sizes, int n_in,
                              void* d_out, int out_size, void* d_ws, size_t ws_size,
                              hipStream_t stream) {
    (void)in_sizes; (void)n_in; (void)out_size; (void)d_ws; (void)ws_size;
    const float*   q         = (const float*)d_in[0];
    const float*   mem_k     = (const float*)d_in[1];
    const float*   mem_v     = (const float*)d_in[2];
    const uint8_t* mem_mask  = (const uint8_t*)d_in[3];
    const float*   local_out = (const float*)d_in[4];
    const float*   scale     = (const float*)d_in[5];
    float*         out       = (float*)d_out;

    dim3 grid(NQ / WAVES);        // 4096 blocks, one query per wave
    dim3 block(WAVES * 32);       // 256 threads = 8 wave32s
    knn_mem_attn_kernel<<<grid, block, 0, stream>>>(q, mem_k, mem_v, mem_mask,
                                                    local_out, scale, out);
}